// Net_14147622273464
// MI455X (gfx1250) — compile-verified
//
#include <hip/hip_runtime.h>
#include <math.h>

typedef float v2f __attribute__((ext_vector_type(2)));
typedef float v8f __attribute__((ext_vector_type(8)));

#define FIN   512
#define FH    16
#define FOUT  40
#define FOUTP 48   // 40 padded to 3 WMMA n-tiles of 16

// ---------------- small elementwise kernels ----------------

__global__ void k_fill(float* __restrict__ p, float v, int n) {
  int i = blockIdx.x * blockDim.x + threadIdx.x;
  if (i < n) p[i] = v;
}

__global__ void k_deg(const int* __restrict__ dst, float* __restrict__ deg, int E) {
  int e = blockIdx.x * blockDim.x + threadIdx.x;
  if (e < E) unsafeAtomicAdd(&deg[dst[e]], 1.0f);
}

__global__ void k_dis(float* __restrict__ deg, int n) {
  int i = blockIdx.x * blockDim.x + threadIdx.x;
  if (i < n) deg[i] = rsqrtf(deg[i]);   // deg >= 1 (self-loop always present)
}

__global__ void k_norm(const int* __restrict__ src, const int* __restrict__ dst,
                       const float* __restrict__ dis, float* __restrict__ norm, int E) {
  int e = blockIdx.x * blockDim.x + threadIdx.x;
  if (e < E) norm[e] = dis[src[e]] * dis[dst[e]];
}

__global__ void k_w2pad(const float* __restrict__ W2, float* __restrict__ W2p) {
  int i = blockIdx.x * blockDim.x + threadIdx.x;
  if (i < FH * FOUTP) {
    int k = i / FOUTP, n = i % FOUTP;
    W2p[i] = (n < FOUT) ? W2[k * FOUT + n] : 0.0f;
  }
}

// A[i] = T[i] * dis[node]^2  (self-loop contribution), F = feature width
__global__ void k_selfinit(const float* __restrict__ T, const float* __restrict__ dis,
                           float* __restrict__ A, int total, int F) {
  int i = blockIdx.x * blockDim.x + threadIdx.x;
  if (i < total) {
    float d = dis[i / F];
    A[i] = T[i] * d * d;
  }
}

__global__ void k_biasrelu(const float* __restrict__ A, const float* __restrict__ b1,
                           float* __restrict__ H, int total) {
  int i = blockIdx.x * blockDim.x + threadIdx.x;
  if (i < total) {
    float v = A[i] + b1[i % FH];
    H[i] = v > 0.0f ? v : 0.0f;
  }
}

// ---------------- WMMA GEMMs (fp32, V_WMMA_F32_16X16X4_F32) ----------------

__global__ __launch_bounds__(256)
void k_gemm1(const float* __restrict__ X, const float* __restrict__ W1,
             float* __restrict__ T1, int nTiles) {
  int wave = (blockIdx.x * blockDim.x + threadIdx.x) >> 5;
  int lane = threadIdx.x & 31;
  if (wave >= nTiles) return;               // wave-uniform: EXEC all-1s at WMMA
  int m  = lane & 15;                       // row within tile == col for B/D frags
  int kh = lane >> 4;                       // which (K pair) half this lane holds
  size_t row0 = (size_t)wave * 16;
  const float* xr = X + (row0 + m) * FIN;
  v8f c = {0.f, 0.f, 0.f, 0.f, 0.f, 0.f, 0.f, 0.f};
#pragma unroll 4
  for (int k0 = 0; k0 < FIN; k0 += 4) {
    int ka = k0 + 2 * kh;
    v2f a = *(const v2f*)(xr + ka);         // A: rows in lanes, K pair in vgprs
    v2f b;
    b.x = W1[ka * FH + m];                  // B: cols in lanes, K pair in vgprs
    b.y = W1[(ka + 1) * FH + m];
    c = __builtin_amdgcn_wmma_f32_16x16x4_f32(false, a, false, b, (short)0, c,
                                              false, false);
  }
#pragma unroll
  for (int r = 0; r < 8; ++r)
    T1[(row0 + r + 8 * kh) * FH + m] = c[r];
}

__global__ __launch_bounds__(256)
void k_gemm2(const float* __restrict__ H, const float* __restrict__ W2p,
             float* __restrict__ T2, int nTiles) {
  int wave = (blockIdx.x * blockDim.x + threadIdx.x) >> 5;
  int lane = threadIdx.x & 31;
  if (wave >= nTiles) return;
  int m  = lane & 15;
  int kh = lane >> 4;
  size_t row0 = (size_t)wave * 16;
  const float* hr = H + (row0 + m) * FH;
  v2f a[4];
#pragma unroll
  for (int t = 0; t < 4; ++t) a[t] = *(const v2f*)(hr + 4 * t + 2 * kh);
#pragma unroll
  for (int nt = 0; nt < 3; ++nt) {
    v8f c = {0.f, 0.f, 0.f, 0.f, 0.f, 0.f, 0.f, 0.f};
#pragma unroll
    for (int t = 0; t < 4; ++t) {
      int ka = 4 * t + 2 * kh;
      v2f b;
      b.x = W2p[ka * FOUTP + nt * 16 + m];
      b.y = W2p[(ka + 1) * FOUTP + nt * 16 + m];
      c = __builtin_amdgcn_wmma_f32_16x16x4_f32(false, a[t], false, b, (short)0, c,
                                                false, false);
    }
    int col = nt * 16 + m;
    if (col < FOUT) {                        // store-only divergence (post-WMMA)
#pragma unroll
      for (int r = 0; r < 8; ++r)
        T2[(row0 + r + 8 * kh) * FOUT + col] = c[r];
    }
  }
}

// ---------------- edge scatter (L2-resident fp32 atomics) ----------------

__global__ void k_scatter16(const int* __restrict__ src, const int* __restrict__ dst,
                            const float* __restrict__ norm, const float* __restrict__ T,
                            float* __restrict__ A, int E) {
  int e = blockIdx.x * blockDim.x + threadIdx.x;
  if (e >= E) return;
  int s = src[e], d = dst[e];
  float w = norm[e];
  const float4* ts = (const float4*)(T + (size_t)s * FH);  // 64B row, aligned
  float* ad = A + (size_t)d * FH;
#pragma unroll
  for (int q = 0; q < 4; ++q) {
    float4 v = ts[q];
    unsafeAtomicAdd(ad + 4 * q + 0, v.x * w);
    unsafeAtomicAdd(ad + 4 * q + 1, v.y * w);
    unsafeAtomicAdd(ad + 4 * q + 2, v.z * w);
    unsafeAtomicAdd(ad + 4 * q + 3, v.w * w);
  }
}

__global__ void k_scatter40(const int* __restrict__ src, const int* __restrict__ dst,
                            const float* __restrict__ norm, const float* __restrict__ T,
                            float* __restrict__ A, int E) {
  int e = blockIdx.x * blockDim.x + threadIdx.x;
  if (e >= E) return;
  int s = src[e], d = dst[e];
  float w = norm[e];
  const float4* ts = (const float4*)(T + (size_t)s * FOUT); // 160B row, 16B aligned
  float* ad = A + (size_t)d * FOUT;
#pragma unroll
  for (int q = 0; q < 10; ++q) {
    float4 v = ts[q];
    unsafeAtomicAdd(ad + 4 * q + 0, v.x * w);
    unsafeAtomicAdd(ad + 4 * q + 1, v.y * w);
    unsafeAtomicAdd(ad + 4 * q + 2, v.z * w);
    unsafeAtomicAdd(ad + 4 * q + 3, v.w * w);
  }
}

// ---------------- bias + log_softmax (in place on d_out) ----------------

__global__ void k_lsm(float* __restrict__ out, const float* __restrict__ b2, int n) {
  int i = blockIdx.x * blockDim.x + threadIdx.x;
  if (i >= n) return;
  float* row = out + (size_t)i * FOUT;
  float v[FOUT];
  float mx = -INFINITY;
#pragma unroll
  for (int c = 0; c < FOUT; ++c) { v[c] = row[c] + b2[c]; mx = fmaxf(mx, v[c]); }
  float s = 0.0f;
#pragma unroll
  for (int c = 0; c < FOUT; ++c) s += expf(v[c] - mx);
  float lse = logf(s) + mx;
#pragma unroll
  for (int c = 0; c < FOUT; ++c) row[c] = v[c] - lse;
}

// ---------------- host-side orchestration ----------------

extern "C" void kernel_launch(void* const* d_in, const int* in_sizes, int n_in,
                              void* d_out, int out_size, void* d_ws, size_t ws_size,
                              hipStream_t stream) {
  const float* x    = (const float*)d_in[0];
  const int*   edge = (const int*)d_in[1];
  const float* W1   = (const float*)d_in[2];
  const float* b1   = (const float*)d_in[3];
  const float* W2   = (const float*)d_in[4];
  const float* b2   = (const float*)d_in[5];

  const int N = in_sizes[0] / FIN;      // 100000
  const int E = in_sizes[1] / 2;        // 3200000
  const int* src = edge;
  const int* dst = edge + E;

  // workspace layout (floats)
  float* ws   = (float*)d_ws;
  float* deg  = ws;                        // N   (becomes dis, in place)
  float* nrm  = deg + N;                   // E
  float* t1   = nrm + E;                   // N*FH   (later reused as h1)
  float* agg1 = t1 + (size_t)N * FH;       // N*FH
  float* t2   = agg1 + (size_t)N * FH;     // N*FOUT
  float* w2p  = t2 + (size_t)N * FOUT;     // FH*FOUTP
  float* out  = (float*)d_out;             // N*FOUT (also layer-2 accumulator)

  const int TB = 256;
  dim3 blk(TB);
  int gN   = (N + TB - 1) / TB;
  int gE   = (E + TB - 1) / TB;
  int gNF  = (N * FH + TB - 1) / TB;
  int gNO  = (N * FOUT + TB - 1) / TB;
  int tiles = (N + 15) / 16;               // 6250 (N % 16 == 0)
  int gG   = (tiles + 7) / 8;              // 8 waves / 256-thread block

  // 1) degree (self-loop = 1) -> dis
  k_fill<<<gN, blk, 0, stream>>>(deg, 1.0f, N);
  k_deg<<<gE, blk, 0, stream>>>(dst, deg, E);
  k_dis<<<gN, blk, 0, stream>>>(deg, N);
  // 2) edge norms
  k_norm<<<gE, blk, 0, stream>>>(src, dst, deg, nrm, E);
  // 3) pad W2 to 16x48
  k_w2pad<<<(FH * FOUTP + TB - 1) / TB, blk, 0, stream>>>(W2, w2p);

  // layer 1: t1 = x @ W1 ; agg1 = selfloop + scatter ; h1 = relu(agg1 + b1)
  k_gemm1<<<gG, blk, 0, stream>>>(x, W1, t1, tiles);
  k_selfinit<<<gNF, blk, 0, stream>>>(t1, deg, agg1, N * FH, FH);
  k_scatter16<<<gE, blk, 0, stream>>>(src, dst, nrm, t1, agg1, E);
  k_biasrelu<<<gNF, blk, 0, stream>>>(agg1, b1, t1, N * FH);   // h1 overwrites t1

  // layer 2: t2 = h1 @ W2 ; out = selfloop + scatter ; log_softmax(out + b2)
  k_gemm2<<<gG, blk, 0, stream>>>(t1, w2p, t2, tiles);
  k_selfinit<<<gNO, blk, 0, stream>>>(t2, deg, out, N * FOUT, FOUT);
  k_scatter40<<<gE, blk, 0, stream>>>(src, dst, nrm, t2, out, E);
  k_lsm<<<gN, blk, 0, stream>>>(out, b2, N);
}